// GlobalAttention_55155970015564
// MI455X (gfx1250) — compile-verified
//
#include <hip/hip_runtime.h>
#include <cstdint>

typedef __attribute__((ext_vector_type(2))) float v2f;
typedef __attribute__((ext_vector_type(8))) float v8f;

#define WAVES 4
#define TPB 128
#define RS 132  // padded row stride (floats): 132 mod 64 = 4 -> conflict-free strided reads

__device__ __forceinline__ void issue_tile_loads(const float* __restrict__ x,
                                                 uint32_t bufLds, int tbase,
                                                 int lane, int N) {
#pragma unroll
  for (int r = 0; r < 16; ++r) {
    int node = tbase + r; node = node < N ? node : (N - 1);  // clamp; masked later
    uint32_t voff = (uint32_t)node * 512u + ((uint32_t)lane << 4);
    uint32_t dst  = bufLds + (uint32_t)(r * RS * 4) + ((uint32_t)lane << 4);
    asm volatile("global_load_async_to_lds_b128 %0, %1, %2"
                 :: "v"(dst), "v"(voff), "s"(x) : "memory");
  }
}

__global__ __launch_bounds__(TPB) void gap_kernel(
    const float* __restrict__ x, const int* __restrict__ batch,
    const float* __restrict__ gate_w, const float* __restrict__ gate_b,
    float* __restrict__ out, int N)
{
  __shared__ __align__(16) float tile[WAVES][2][16 * RS];  // double-buffered per wave
  __shared__ float gbuf[WAVES][16];
  __shared__ float wlds[128];
  __shared__ float accb[WAVES][128];
  __shared__ float mbuf[WAVES];
  __shared__ float sbuf[WAVES];

  const int tid  = threadIdx.x;
  const int wave = tid >> 5;
  const int lane = tid & 31;
  const int g    = blockIdx.x;

  wlds[tid] = gate_w[tid];
  const float bias = gate_b[0];

  // binary search segment bounds in sorted batch (uniform per block -> scalarized)
  int lo = 0, hi = N;
  while (lo < hi) { int mid = (lo + hi) >> 1; if (batch[mid] < g) lo = mid + 1; else hi = mid; }
  const int segStart = lo;
  hi = N;
  while (lo < hi) { int mid = (lo + hi) >> 1; if (batch[mid] <= g) lo = mid + 1; else hi = mid; }
  const int segEnd = lo;
  const int count = segEnd - segStart;

  if (count <= 0) {            // empty graph -> zero row (reference's -inf guard)
    out[(size_t)g * 128 + tid] = 0.0f;
    return;
  }
  __syncthreads();             // wlds visible to all waves

  const int m16   = lane & 15;
  const int khalf = lane >> 4;
  float m = -__builtin_inff();
  float s = 0.0f;
  float a0 = 0.f, a1 = 0.f, a2 = 0.f, a3 = 0.f;   // features 4*lane .. 4*lane+3

  float* buf0 = tile[wave][0];
  float* buf1 = tile[wave][1];
  const uint32_t lds0 = (uint32_t)(uintptr_t)buf0;  // low 32 bits = LDS offset
  const uint32_t lds1 = (uint32_t)(uintptr_t)buf1;
  const int numTiles = (count + 15) >> 4;

  // ---- software pipeline: prefetch tile t+1 while computing tile t ----
  int cur = 0;
  if (wave < numTiles)
    issue_tile_loads(x, lds0, segStart + (wave << 4), lane, N);

  for (int t = wave; t < numTiles; t += WAVES) {
    const int  tnext    = t + WAVES;
    const bool havenext = tnext < numTiles;

    if (havenext)  // issue next tile into the other buffer (ASYNCcnt -> 32)
      issue_tile_loads(x, cur ? lds0 : lds1, segStart + (tnext << 4), lane, N);

    // async loads complete in order: <=16 outstanding means current tile landed
    if (havenext) asm volatile("s_wait_asynccnt 16" ::: "memory");
    else          asm volatile("s_wait_asynccnt 0"  ::: "memory");

    float* mytile = cur ? buf1 : buf0;
    const int tbase  = segStart + (t << 4);
    const int tcount = min(16, segEnd - tbase);

    // ---- 16 gates via V_WMMA_F32_16X16X4_F32, K-loop over 32 chunks of 4 ----
    // A[mrow][k] = x_tile[mrow][4kk+k]; B[k][n] = w[4kk+k] for all n
    // => every column of D is the gate vector.
    v8f c = {0.f,0.f,0.f,0.f,0.f,0.f,0.f,0.f};
    for (int kk = 0; kk < 32; ++kk) {
      const int off = (kk << 2) + (khalf << 1);
      v2f a, b;
      a.x = mytile[m16 * RS + off];
      a.y = mytile[m16 * RS + off + 1];
      b.x = wlds[off];
      b.y = wlds[off + 1];
      c = __builtin_amdgcn_wmma_f32_16x16x4_f32(false, a, false, b,
                                                (short)0, c, false, false);
    }

    // publish gates (lane0: rows 0-7, lane16: rows 8-15), broadcast via LDS
    if (lane == 0 || lane == 16) {
#pragma unroll
      for (int v = 0; v < 8; ++v) gbuf[wave][khalf * 8 + v] = c[v] + bias;
    }
    asm volatile("s_wait_dscnt 0" ::: "memory");

    // ---- online softmax update (redundant per lane, uniform) ----
    float tmax = -__builtin_inff();
    float gv[16];
#pragma unroll
    for (int n = 0; n < 16; ++n) {
      gv[n] = gbuf[wave][n];
      if (n < tcount) tmax = fmaxf(tmax, gv[n]);
    }
    const float mnew  = fmaxf(m, tmax);
    const float scale = __expf(m - mnew);   // first tile: exp(-inf)=0
    float p[16]; float psum = 0.f;
#pragma unroll
    for (int n = 0; n < 16; ++n) {
      float pv = (n < tcount) ? __expf(gv[n] - mnew) : 0.f;
      p[n] = pv; psum += pv;
    }
    s = s * scale + psum;
    a0 *= scale; a1 *= scale; a2 *= scale; a3 *= scale;
    m = mnew;

    // ---- weighted accumulation: acc += p[n] * x_tile[n][4*lane..] ----
#pragma unroll
    for (int n = 0; n < 16; ++n) {
      const float* rp = mytile + n * RS + (lane << 2);
      a0 += p[n] * rp[0];
      a1 += p[n] * rp[1];
      a2 += p[n] * rp[2];
      a3 += p[n] * rp[3];
    }

    cur ^= 1;
  }

  // ---- cross-wave combine ----
  if (lane == 0) { mbuf[wave] = m; sbuf[wave] = s; }
  {
    float* ab = accb[wave] + (lane << 2);
    ab[0] = a0; ab[1] = a1; ab[2] = a2; ab[3] = a3;
  }
  __syncthreads();

  {
    float mstar = -__builtin_inff();
#pragma unroll
    for (int w = 0; w < WAVES; ++w) mstar = fmaxf(mstar, mbuf[w]);
    float num = 0.f, den = 0.f;
#pragma unroll
    for (int w = 0; w < WAVES; ++w) {
      const float al = __expf(mbuf[w] - mstar);   // idle wave: exp(-inf)=0
      num += al * accb[w][tid];
      den += al * sbuf[w];
    }
    out[(size_t)g * 128 + tid] = num / (den + 1e-16f);
  }
}

extern "C" void kernel_launch(void* const* d_in, const int* in_sizes, int n_in,
                              void* d_out, int out_size, void* d_ws, size_t ws_size,
                              hipStream_t stream) {
  (void)n_in; (void)d_ws; (void)ws_size;
  const float* x      = (const float*)d_in[0];
  const int*   batch  = (const int*)d_in[1];
  // d_in[2] = num_graphs scalar (derive G from out_size instead)
  const float* gate_w = (const float*)d_in[3];
  const float* gate_b = (const float*)d_in[4];
  const int N = in_sizes[0] / 128;
  const int G = out_size / 128;
  gap_kernel<<<G, TPB, 0, stream>>>(x, batch, gate_w, gate_b, (float*)d_out, N);
}